// KnowledgeEmbedding_8907762172017
// MI455X (gfx1250) — compile-verified
//
#include <hip/hip_runtime.h>

#define EMBED   100
#define EPAD    104
#define BATCH   4096
#define NUM_NEG 256
#define NRELS   8
#define MT      16          // M-tile (batch rows per block)
#define THREADS 128         // 4 wave32s
#define KSTEPS  25          // K=100 / 4 per fp32 WMMA issue

typedef __attribute__((ext_vector_type(2))) float v2f;
typedef __attribute__((ext_vector_type(8))) float v8f;

struct RelParams {
    const float* head[NRELS];
    const float* tail[NRELS];
    const float* bias[NRELS];
    int hcol[NRELS];
    int tcol[NRELS];
};

// numerically stable  log(1 + exp(x))  ==  -log_sigmoid(-x)
__device__ __forceinline__ float softplus_f(float x) {
    return fmaxf(x, 0.0f) + __logf(1.0f + __expf(-fabsf(x)));
}

__global__ void kg_zero_out(float* out) { out[0] = 0.0f; }

__global__ __launch_bounds__(THREADS) void kg_loss_kernel(
    RelParams rp, const float* __restrict__ rel_vecs,
    const int* __restrict__ batch_idxs, const int* __restrict__ neg_idxs,
    float* __restrict__ out)
{
    const int r     = blockIdx.y;          // relation 0..7
    const int mtile = blockIdx.x;          // 0..255 -> batch rows [mtile*16, +16)
    const int tid   = threadIdx.x;
    const int lane  = tid & 31;
    const int wave  = tid >> 5;            // 0..3

    __shared__ float exs[MT][EPAD];        // head[h] + rel_vec
    __shared__ float tvs[MT][EPAD];        // tail[t]
    __shared__ float tbias[MT];
    __shared__ float red[THREADS];

    const float* __restrict__ headT = rp.head[r];
    const float* __restrict__ tailT = rp.tail[r];
    const float* __restrict__ biasT = rp.bias[r];
    const int hcol = rp.hcol[r];
    const int tcol = rp.tcol[r];
    const float* __restrict__ rv = rel_vecs + r * EMBED;

    // ---- stage ex / tv tiles into LDS (gathers) ----
    for (int i = tid; i < MT * EMBED; i += THREADS) {
        const int row = i / EMBED;
        const int e   = i % EMBED;
        const int b   = mtile * MT + row;
        const int h   = batch_idxs[b * 8 + hcol];
        const int t   = batch_idxs[b * 8 + tcol];
        exs[row][e] = headT[(long)h * EMBED + e] + rv[e];
        tvs[row][e] = tailT[(long)t * EMBED + e];
    }
    if (tid < MT) {
        const int b = mtile * MT + tid;
        const int t = batch_idxs[b * 8 + tcol];
        tbias[tid] = biasT[t];
    }
    __syncthreads();

    float lsum = 0.0f;

    // ---- positive term: 16 rowwise dots (ex . tv) + bias ----
    if (tid < MT) {
        float acc = 0.0f;
        #pragma unroll 4
        for (int e = 0; e < EMBED; ++e)
            acc = fmaf(exs[tid][e], tvs[tid][e], acc);
        lsum += softplus_f(-(acc + tbias[tid]));   // -log_sigmoid(pos)
    }

    // ---- negative term: neg = ex(16x100) @ nv^T(100x16), fp32 WMMA ----
    // A frag (16x4 f32): lane L<16 holds M=L, K={k0,k0+1}; lane>=16 holds M=L-16, K={k0+2,k0+3}
    // B frag (4x16 f32): lane L<16 holds N=L, K={k0,k0+1}; lane>=16 holds N=L-16, K={k0+2,k0+3}
    const int mrow  = lane & 15;
    const int khalf = (lane >> 4) << 1;            // 0 or 2
    const int mbase = (lane >> 4) << 3;            // D rows: i (+8 for upper half)

    // Preload ALL A fragments for this wave's M-tile into registers (tile-invariant),
    // and the 8 per-row bias values this lane needs.
    v2f afr[KSTEPS];
    #pragma unroll
    for (int kk = 0; kk < KSTEPS; ++kk) {
        afr[kk].x = exs[mrow][4 * kk + khalf];
        afr[kk].y = exs[mrow][4 * kk + khalf + 1];
    }
    float bb[8];
    #pragma unroll
    for (int i = 0; i < 8; ++i) bb[i] = tbias[mbase + i];

    // Each wave owns N-tiles {wave, wave+4, wave+8, wave+12}; process as 2 pairs
    // -> two independent WMMA accumulator chains + two independent load streams.
    #pragma unroll
    for (int pair = 0; pair < 2; ++pair) {
        const int t0 = wave + pair * 8;            // N-tile A
        const int t1 = t0 + 4;                     // N-tile B
        const int n0 = t0 * 16 + (lane & 15);
        const int n1 = t1 * 16 + (lane & 15);
        const int i0 = neg_idxs[r * NUM_NEG + n0];
        const int i1 = neg_idxs[r * NUM_NEG + n1];
        const float* __restrict__ p0 = tailT + (long)i0 * EMBED + khalf;
        const float* __restrict__ p1 = tailT + (long)i1 * EMBED + khalf;

        v8f acc0 = {}, acc1 = {};
        #pragma unroll
        for (int kk = 0; kk < KSTEPS; ++kk) {
            const v2f b0 = *(const v2f*)(p0 + 4 * kk);   // 8B-aligned global_load_b64
            const v2f b1 = *(const v2f*)(p1 + 4 * kk);
            acc0 = __builtin_amdgcn_wmma_f32_16x16x4_f32(
                false, afr[kk], false, b0, (short)0, acc0, false, false);
            acc1 = __builtin_amdgcn_wmma_f32_16x16x4_f32(
                false, afr[kk], false, b1, (short)0, acc1, false, false);
        }
        #pragma unroll
        for (int i = 0; i < 8; ++i) {
            lsum += softplus_f(acc0[i] + bb[i]);   // -log_sigmoid(-neg)
            lsum += softplus_f(acc1[i] + bb[i]);
        }
    }

    // ---- block reduction + scalar accumulation ----
    red[tid] = lsum;
    __syncthreads();
    for (int s = THREADS / 2; s > 0; s >>= 1) {
        if (tid < s) red[tid] += red[tid + s];
        __syncthreads();
    }
    if (tid == 0) atomicAdd(out, red[0] * (1.0f / (float)BATCH));
}

extern "C" void kernel_launch(void* const* d_in, const int* in_sizes, int n_in,
                              void* d_out, int out_size, void* d_ws, size_t ws_size,
                              hipStream_t stream) {
    const float* user_emb      = (const float*)d_in[0];
    const float* product_emb   = (const float*)d_in[1];
    const float* word_emb      = (const float*)d_in[2];
    const float* brand_emb     = (const float*)d_in[3];
    const float* category_emb  = (const float*)d_in[4];
    const float* rproduct_emb  = (const float*)d_in[5];
    const float* rel_vecs      = (const float*)d_in[6];
    const float* purchase_b    = (const float*)d_in[7];
    const float* mentions_b    = (const float*)d_in[8];
    const float* describe_b    = (const float*)d_in[9];
    const float* produced_b    = (const float*)d_in[10];
    const float* belongs_b     = (const float*)d_in[11];
    const float* also_bought_b = (const float*)d_in[12];
    const float* also_viewed_b = (const float*)d_in[13];
    const float* together_b    = (const float*)d_in[14];
    const int*   batch_idxs    = (const int*)d_in[15];
    const int*   neg_idxs      = (const int*)d_in[16];

    RelParams rp;
    const float* heads[NRELS] = { user_emb, user_emb, product_emb, product_emb,
                                  product_emb, product_emb, product_emb, product_emb };
    const float* tails[NRELS] = { product_emb, word_emb, word_emb, brand_emb,
                                  category_emb, rproduct_emb, rproduct_emb, rproduct_emb };
    const float* biases[NRELS] = { purchase_b, mentions_b, describe_b, produced_b,
                                   belongs_b, also_bought_b, also_viewed_b, together_b };
    const int hcols[NRELS] = { 0, 0, 1, 1, 1, 1, 1, 1 };
    const int tcols[NRELS] = { 1, 2, 2, 3, 4, 5, 6, 7 };
    for (int i = 0; i < NRELS; ++i) {
        rp.head[i] = heads[i];
        rp.tail[i] = tails[i];
        rp.bias[i] = biases[i];
        rp.hcol[i] = hcols[i];
        rp.tcol[i] = tcols[i];
    }

    kg_zero_out<<<dim3(1), dim3(1), 0, stream>>>((float*)d_out);

    dim3 grid(BATCH / MT, NRELS);
    kg_loss_kernel<<<grid, dim3(THREADS), 0, stream>>>(
        rp, rel_vecs, batch_idxs, neg_idxs, (float*)d_out);
}